// SparkNetAlpha_19997367730513
// MI455X (gfx1250) — compile-verified
//
#include <hip/hip_runtime.h>

typedef __attribute__((ext_vector_type(2))) float v2f;
typedef __attribute__((ext_vector_type(8))) float v8f;

#define NN 8192
#define KK 256
#define KSEG 4
#define KLEN (NN / KSEG)   // 2048

// ---------------- deterministic hash RNG (stand-in for JAX threefry) -------
__device__ __forceinline__ unsigned hash_u32(unsigned x) {
  x ^= x >> 17u; x *= 0xed5ad4bbu;
  x ^= x >> 11u; x *= 0xac4c1b51u;
  x ^= x >> 15u; x *= 0x31848babu;
  x ^= x >> 14u;
  return x;
}
__device__ __forceinline__ float u01(unsigned x) {
  // strictly inside (0,1): safe for logf
  return ((float)(hash_u32(x) >> 8) + 0.5f) * (1.0f / 16777216.0f);
}
__device__ __forceinline__ float gumbelf(unsigned x) {
  return -__logf(-__logf(u01(x)));
}

// ---------------------------------------------------------------------------
// Pass 1: fused  y_part[seg] += W_tile @ s_tile  (fp32 WMMA 16x16x4, B bcast)
//         and    W_out = clip(W * (1-LR_GLOBAL_DECAY), -1, 1)
// HBM-bound: 256MB read + 256MB write, one pass over W.
// Each wave owns 16 rows; lane l (l<16) loads cols {k,k+1}, lane l+16 loads
// cols {k+2,k+3} of the same row -> matches the ISA 16x4 fp32 A layout and
// gives contiguous 16B/row per step. B holds s[k..k+3] replicated across all
// 16 columns, so every column of the 16x16 accumulator equals y for that row.
// ---------------------------------------------------------------------------
__global__ __launch_bounds__(256) void k1_gemv_decay(
    const float* __restrict__ W, const float* __restrict__ s,
    float* __restrict__ Wout, float* __restrict__ ypart) {
  __shared__ float s_sh[KLEN];
  const int seg  = blockIdx.x;       // K segment (0..3)
  const int rb   = blockIdx.y;       // row block (0..63), 128 rows each
  const int tid  = threadIdx.x;
  const int lane = tid & 31;
  const int wave = tid >> 5;
  const int k0   = seg * KLEN;

  for (int i = tid; i < KLEN; i += 256) s_sh[i] = s[k0 + i];
  __syncthreads();

  const int half = lane >> 4;                       // 0: K%4 in {0,1}, 1: {2,3}
  const int mrow = rb * 128 + wave * 16 + (lane & 15);
  const size_t rowOff = (size_t)mrow * NN;
  const float* wp = W    + rowOff + k0 + half * 2;
  float*       wo = Wout + rowOff + k0 + half * 2;

  v8f acc = {};
  for (int k = 0; k < KLEN; k += 4) {
    v2f a = *(const v2f*)(wp + k);                  // A: (M=mrow, K=k..k+1 | k+2..k+3)
    __builtin_prefetch(wp + k + 64, 0, 1);          // global_prefetch_b8, stream ahead
    v2f b;                                          // B: row K broadcast over N
    b.x = s_sh[k + half * 2];
    b.y = s_sh[k + half * 2 + 1];
    // fused global decay + clip store (scan kernel patches the <=256 scattered
    // elements afterwards with the exact post-scatter decayed value)
    v2f wd;
    wd.x = fminf(fmaxf(a.x * 0.999f, -1.0f), 1.0f);
    wd.y = fminf(fmaxf(a.y * 0.999f, -1.0f), 1.0f);
    *(v2f*)(wo + k) = wd;
    acc = __builtin_amdgcn_wmma_f32_16x16x4_f32(
        false, a, false, b, (short)0, acc, false, false);
  }
  // D layout: acc[r] @ lane<16 is (M=r, N=lane); @ lane>=16 is (M=r+8).
  // All N columns identical (B broadcast) -> lanes 0 and 16 hold y for 16 rows.
  if ((lane & 15) == 0) {
    float* yp = ypart + (size_t)seg * NN + (rb * 128 + wave * 16 + half * 8);
#pragma unroll
    for (int r = 0; r < 8; ++r) yp[r] = acc[r];
  }
}

// ---------------------------------------------------------------------------
// Pass 2: single-workgroup sequential spark scan (inherent K=256 dependency).
// Prologue: reduce 4 partials -> s = sigmoid(0.95*y + noise), forced sparks,
//           M = M*0.92.  s and M live in dynamic LDS (72KB < 320KB WGP LDS).
// Per step: row read with latest-wins scatter overrides (compacted per step),
//           Gumbel-argmax categorical sampling, scatter updates with decay
//           folded into the W_out write.
// ---------------------------------------------------------------------------
__global__ __launch_bounds__(256) void k2_scan(
    const float* __restrict__ W, const float* __restrict__ M_in,
    const int* __restrict__ pos_in, const float* __restrict__ en_in,
    const int* __restrict__ age_in, const float* __restrict__ ypart,
    float* __restrict__ Wout, int* __restrict__ pos_out,
    float* __restrict__ s_out, float* __restrict__ M_out,
    float* __restrict__ en_out, int* __restrict__ age_out) {
  extern __shared__ float smem[];
  float* s_sh  = smem;                    // NN floats
  float* M_sh  = s_sh + NN;               // NN floats
  float* red_v = M_sh + NN;               // 256
  int*   red_i = (int*)(red_v + 256);     // 256
  int*   sc_row = red_i + 256;            // KK scatter records (row,col,val)
  int*   sc_col = sc_row + KK;
  float* sc_val = (float*)(sc_col + KK);
  int*   m_col  = (int*)(sc_val + KK);    // per-step compacted matches
  int*   m_ord  = m_col + KK;
  float* m_val  = (float*)(m_ord + KK);
  __shared__ int n_sc, m_cnt;

  const int tid = threadIdx.x;

  // ---- prologue ----
  for (int i = tid; i < NN; i += 256) {
    float y = ypart[i] + ypart[NN + i] + ypart[2 * NN + i] + ypart[3 * NN + i];
    float u1 = u01(0x1234567u ^ (unsigned)i);
    float u2 = u01(0x89abcdefu ^ (unsigned)(i * 0x9e3779b9u));
    float nrm = sqrtf(fmaxf(-2.0f * __logf(u1), 0.0f)) * __cosf(6.2831853f * u2);
    float v = 0.95f * y + 0.05f * nrm;          // STATE_DECAY folded into y
    s_sh[i] = 1.0f / (1.0f + __expf(-v));
    M_sh[i] = M_in[i] * 0.92f;                  // MEM_DECAY
  }
  if (tid == 0) n_sc = 0;
  __syncthreads();
  if (tid < KK && age_in[tid] < 5) s_sh[pos_in[tid]] = 1.0f;  // benign race: all write 1.0
  __syncthreads();

  // ---- sequential spark loop ----
  for (int t = 0; t < KK; ++t) {
    const int prev = pos_in[t];

    if (tid == 0) m_cnt = 0;
    __syncthreads();
    const int ns = n_sc;
    if (tid < ns && sc_row[tid] == prev) {      // compact overrides hitting this row
      int p = atomicAdd(&m_cnt, 1);
      m_col[p] = sc_col[tid]; m_ord[p] = tid; m_val[p] = sc_val[tid];
    }
    __syncthreads();
    const int nm = m_cnt;

    float best = -1e30f, best2 = -1e30f; int bi = 0, bi2 = 0;
    for (int j = 0; j < NN / 256; ++j) {
      int c = tid + j * 256;
      float w = W[(size_t)prev * NN + c];
      int bo = -1;                              // latest-wins override
      for (int q = 0; q < nm; ++q)
        if (m_col[q] == c && m_ord[q] > bo) { bo = m_ord[q]; w = m_val[q]; }
      float logit = (fmaxf(w, 0.0f) + 1e-6f) * (1.0f / 0.3f) + 0.8f * M_sh[c];
      float g1 = gumbelf((unsigned)(t * NN + c) ^ 0x0051a7eu);
      float v1 = logit + g1;
      if (v1 > best || (v1 == best && c < bi)) { best = v1; bi = c; }
      float g2 = gumbelf((unsigned)(t * NN + c) ^ 0x7e57e90u);
      float v2 = 2.0f * M_sh[c] + g2;
      if (v2 > best2 || (v2 == best2 && c < bi2)) { best2 = v2; bi2 = c; }
    }

    // argmax reduction 1: transition sample
    red_v[tid] = best; red_i[tid] = bi;
    __syncthreads();
    for (int off = 128; off > 0; off >>= 1) {
      if (tid < off) {
        float ov = red_v[tid + off]; int oi = red_i[tid + off];
        if (ov > red_v[tid] || (ov == red_v[tid] && oi < red_i[tid])) {
          red_v[tid] = ov; red_i[tid] = oi;
        }
      }
      __syncthreads();
    }
    const int samp = red_i[0];
    __syncthreads();
    // argmax reduction 2: memory respawn sample
    red_v[tid] = best2; red_i[tid] = bi2;
    __syncthreads();
    for (int off = 128; off > 0; off >>= 1) {
      if (tid < off) {
        float ov = red_v[tid + off]; int oi = red_i[tid + off];
        if (ov > red_v[tid] || (ov == red_v[tid] && oi < red_i[tid])) {
          red_v[tid] = ov; red_i[tid] = oi;
        }
      }
      __syncthreads();
    }
    const int memp = red_i[0];

    if (tid == 0) {
      bool explore = u01(0xc0ffee1u ^ (unsigned)t) < 0.05f;
      int randp = (int)(hash_u32(0xdeadbe3u ^ (unsigned)t) & (NN - 1));
      int next = explore ? randp : samp;
      size_t idx = (size_t)next * NN + prev;
      float wold = W[idx];
      for (int q = 0; q < ns; ++q)              // latest-wins on (next, prev)
        if (sc_row[q] == next && sc_col[q] == prev) wold = sc_val[q];
      float v = wold * 0.95f + s_sh[prev] * 0.05f;      // LR_EDGE
      sc_row[ns] = next; sc_col[ns] = prev; sc_val[ns] = v; n_sc = ns + 1;
      Wout[idx] = fminf(fmaxf(v * 0.999f, -1.0f), 1.0f); // fix up fused decay
      M_sh[next] += 0.15f;                               // MEM_DEPOSIT
      float e = en_in[t] * 0.98f;                        // SPARK_ENERGY_DECAY
      s_sh[next] = e;
      bool respawn = e < 0.05f;   // M.sum() > 0 always: M >= 0, deposits > 0
      pos_out[t] = respawn ? memp : next;
      en_out[t]  = respawn ? 1.0f : e;
      age_out[t] = respawn ? 0 : (age_in[t] + 1);
    }
    __syncthreads();
  }

  for (int i = tid; i < NN; i += 256) { s_out[i] = s_sh[i]; M_out[i] = M_sh[i]; }
}

// ---------------------------------------------------------------------------
extern "C" void kernel_launch(void* const* d_in, const int* in_sizes, int n_in,
                              void* d_out, int out_size, void* d_ws, size_t ws_size,
                              hipStream_t stream) {
  const float* W   = (const float*)d_in[0];
  const float* s   = (const float*)d_in[1];
  const float* M   = (const float*)d_in[2];
  const int*   sp  = (const int*)d_in[3];
  const float* se  = (const float*)d_in[4];
  const int*   sa  = (const int*)d_in[5];

  // outputs concatenated in return order: pos, s, M, W, energy, age (all 4B)
  int*   pos_out = (int*)d_out;
  float* s_out   = (float*)d_out + KK;
  float* M_out   = s_out + NN;
  float* W_out   = M_out + NN;
  float* en_out  = W_out + (size_t)NN * NN;
  int*   age_out = (int*)(en_out + KK);

  float* ypart = (float*)d_ws;  // KSEG * NN floats = 128KB partial sums

  dim3 g1(KSEG, NN / 128);
  k1_gemv_decay<<<g1, 256, 0, stream>>>(W, s, W_out, ypart);

  // dynamic LDS: s(32K) + M(32K) + red(2K) + scatter(3K) + matches(3K) = 72KB
  const size_t smem = (size_t)(2 * NN + 512) * 4 + (size_t)6 * KK * 4;
  k2_scan<<<1, 256, smem, stream>>>(W, M, sp, se, sa, ypart, W_out,
                                    pos_out, s_out, M_out, en_out, age_out);
}